// ContinuousActionHead_15032385536006
// MI455X (gfx1250) — compile-verified
//
#include <hip/hip_runtime.h>
#include <hip/hip_bf16.h>

typedef __attribute__((ext_vector_type(2))) float v2f;
typedef __attribute__((ext_vector_type(8))) float v8f;

#define D_MODEL 512
#define INT64_MAX_F 9223372036854775807.0f

// digamma for x >= 1 (here x = z^2 + 1 in [1, ~few]): shift to >= 6, asymptotic series.
__device__ __forceinline__ float digammaf_dev(float x) {
    float r = 0.0f;
#pragma unroll
    for (int i = 0; i < 6; ++i) {
        if (x < 6.0f) { r -= 1.0f / x; x += 1.0f; }
    }
    float inv  = 1.0f / x;
    float inv2 = inv * inv;
    // 1/(12x^2) - 1/(120x^4) + 1/(252x^6)
    float ser = inv2 * (0.08333333333f - inv2 * (0.008333333333f - inv2 * 0.003968253968f));
    return r + logf(x) - 0.5f * inv - ser;
}

__global__ __launch_bounds__(256)
void ContinuousActionHead_kernel(const float* __restrict__ x,
                                 const int*   __restrict__ actors,
                                 const float* __restrict__ w,
                                 const float* __restrict__ bvec,
                                 const int*   __restrict__ prev,
                                 float*       __restrict__ out,
                                 int N) {
    __shared__ float lds_w[2 * D_MODEL];   // w staged: [col][k]
    __shared__ float zbuf[8][32];          // per-wave (z0,z1) extraction buffer

    const int tid  = threadIdx.x;
    const int lane = tid & 31;
    const int wv   = tid >> 5;

    // Stage w (2x512 floats) into LDS, uniform across the block.
    for (int i = tid; i < 2 * D_MODEL; i += 256) lds_w[i] = w[i];
    __syncthreads();

    const int tile = blockIdx.x * 8 + wv;   // one wave per 16-actor tile
    const int base = tile * 16;
    if (base >= N) return;                   // wave-uniform exit

    // A-matrix layout for V_WMMA_F32_16X16X4_F32 (16x4, fp32):
    //   lanes 0-15 : row M = lane,    VGPR0=K0, VGPR1=K1
    //   lanes 16-31: row M = lane-16, VGPR0=K2, VGPR1=K3
    const int m  = lane & 15;
    const int kb = (lane >> 4) * 2;

    const int row = actors[base + m];
    const float* arow = x + (long long)row * D_MODEL + kb;

    // B-matrix (4x16): B[k][n] = w[n][k] for n<2 else 0.
    //   lanes 0-15 : K offset 0 ; lanes 16-31 : K offset 2 (same kb as A).
    const int  n       = m;
    const bool valid_n = (n < 2);
    const float* wrow  = lds_w + (valid_n ? n * D_MODEL : 0) + kb;

    v8f c = {};   // 16x16 fp32 accumulator, columns 0/1 = z0,z1 per actor row

#pragma unroll 4
    for (int k = 0; k < D_MODEL; k += 4) {
        v2f a = *(const v2f*)(arow + k);            // global_load_b64 (gathered row)
        v2f bl = *(const v2f*)(wrow + k);           // ds_load_b64
        v2f bm;
        bm.x = valid_n ? bl.x : 0.0f;               // select, no divergence
        bm.y = valid_n ? bl.y : 0.0f;
        c = __builtin_amdgcn_wmma_f32_16x16x4_f32(
                /*neg_a=*/false, a, /*neg_b=*/false, bm,
                /*c_mod=*/(short)0, c, /*reuse_a=*/false, /*reuse_b=*/false);
    }

    // C layout: VGPR r, lanes 0-15 -> (M=r, N=lane); lanes 16-31 -> (M=8+r, N=lane-16).
    // We need N=0,1 for all M: held by lanes 0,1 (M=0..7) and lanes 16,17 (M=8..15).
    if (lane == 0) {
#pragma unroll
        for (int r = 0; r < 8; ++r) zbuf[wv][2 * r + 0] = c[r];
    }
    if (lane == 1) {
#pragma unroll
        for (int r = 0; r < 8; ++r) zbuf[wv][2 * r + 1] = c[r];
    }
    if (lane == 16) {
#pragma unroll
        for (int r = 0; r < 8; ++r) zbuf[wv][16 + 2 * r + 0] = c[r];
    }
    if (lane == 17) {
#pragma unroll
        for (int r = 0; r < 8; ++r) zbuf[wv][16 + 2 * r + 1] = c[r];
    }
    __syncthreads();

    // Epilogue: lanes 0-15 each own one actor.
    if (lane < 16) {
        const int idx = base + lane;
        if (idx < N) {
            const float b0 = bvec[0], b1 = bvec[1];
            float z0 = zbuf[wv][2 * lane + 0] + b0;
            float z1 = zbuf[wv][2 * lane + 1] + b1;
            float A = z0 * z0 + 1.0f;          // logits[:,0]
            float B = z1 * z1 + 1.0f;          // logits[:,1]

            float p      = (float)prev[idx];
            float action = p / INT64_MAX_F;

            float lga  = lgammaf(A);
            float lgb  = lgammaf(B);
            float lgab = lgammaf(A + B);
            float lbeta = lga + lgb - lgab;

            float logprob = (A - 1.0f) * logf(action)
                          + (B - 1.0f) * log1pf(-action)
                          - lbeta;

            float dga  = digammaf_dev(A);
            float dgb  = digammaf_dev(B);
            float dgab = digammaf_dev(A + B);
            float entropy = lbeta - (A - 1.0f) * dga - (B - 1.0f) * dgb
                          + (A + B - 2.0f) * dgab;

            out[idx]                 = action * INT64_MAX_F;   // action_return
            out[N + idx]             = logprob;                // logprob
            out[2 * N + idx]         = entropy;                // entropy
            out[3 * N + 2 * idx + 0] = A;                      // logits interleaved
            out[3 * N + 2 * idx + 1] = B;
        }
    }
}

extern "C" void kernel_launch(void* const* d_in, const int* in_sizes, int n_in,
                              void* d_out, int out_size, void* d_ws, size_t ws_size,
                              hipStream_t stream) {
    const float* x      = (const float*)d_in[0];
    const int*   actors = (const int*)  d_in[1];
    const float* w      = (const float*)d_in[2];
    const float* b      = (const float*)d_in[3];
    const int*   prev   = (const int*)  d_in[4];
    float*       out    = (float*)d_out;

    const int N     = in_sizes[1];          // number of actors
    const int tiles = (N + 15) / 16;        // one wave per 16 actors
    const int grid  = (tiles + 7) / 8;      // 8 waves (256 threads) per block

    ContinuousActionHead_kernel<<<grid, 256, 0, stream>>>(x, actors, w, b, prev, out, N);
}